// SConv2dSubsampling_48507360641265
// MI455X (gfx1250) — compile-verified
//
#include <hip/hip_runtime.h>

typedef __attribute__((ext_vector_type(2))) float v2f;
typedef __attribute__((ext_vector_type(8))) float v8f;

#define B_   16
#define D_   80
#define L_   1024
#define C_   256
#define OH_  512
#define OW_  40
#define PPB  (OH_ * OW_)          /* 20480 positions per batch image  */
#define TPB_ (PPB / 16)           /* 1280 position-tiles per batch    */

__device__ __forceinline__ float img_load(const float* __restrict__ x,
                                          int b, int ih, int iw) {
    // image(b, ih, iw) = x[b][iw][ih]  (x is (B, D, L)); zero-pad below 0.
    // Upper bounds (ih<=1023, iw<=79) can never be exceeded for this conv.
    bool ok = (ih >= 0) && (iw >= 0);
    return ok ? x[b * (D_ * L_) + iw * L_ + ih] : 0.0f;
}

// ---------------------------------------------------------------------------
// Kernel 1: conv(1->256, 3x3, s=2, p=1) as WMMA GEMM with K=9 (3 x K=4 blocks)
// One wave computes a 16-position x 256-channel stripe (16 channel tiles).
// A = weights (M=16 channels x K=4), B = im2col patches (K=4 x N=16 positions)
// D tile: VGPR r, lanes 0-15 -> channel r (N=pos 0..15), lanes 16-31 -> r+8.
// ---------------------------------------------------------------------------
__global__ void __launch_bounds__(256)
conv_wmma_kernel(const float* __restrict__ x, const float* __restrict__ w,
                 float* __restrict__ out) {
    const int lane = threadIdx.x & 31;
    const int wave = blockIdx.x * (blockDim.x >> 5) + (threadIdx.x >> 5);
    const int b    = wave / TPB_;
    const int tp   = wave % TPB_;
    const int pbase = tp * 16;
    const int n    = lane & 15;
    const int half = lane >> 4;

    const int p   = pbase + n;
    const int oh  = p / OW_;
    const int ow  = p % OW_;
    const int ih0 = 2 * oh - 1;
    const int iw0 = 2 * ow - 1;

    // ISA A/B VGPR layout: lanes 0-15 carry K={4i,4i+1}, lanes 16-31 K={4i+2,4i+3}.
    // k = kh*3+kw. half 0 needs k={0,1,4,5,8}; half 1 needs k={2,3,6,7}.
    float s0, s1, s2, s3, s4;
    if (half == 0) {
        s0 = img_load(x, b, ih0 + 0, iw0 + 0);   // k0
        s1 = img_load(x, b, ih0 + 0, iw0 + 1);   // k1
        s2 = img_load(x, b, ih0 + 1, iw0 + 1);   // k4
        s3 = img_load(x, b, ih0 + 1, iw0 + 2);   // k5
        s4 = img_load(x, b, ih0 + 2, iw0 + 2);   // k8
    } else {
        s0 = img_load(x, b, ih0 + 0, iw0 + 2);   // k2
        s1 = img_load(x, b, ih0 + 1, iw0 + 0);   // k3
        s2 = img_load(x, b, ih0 + 2, iw0 + 0);   // k6
        s3 = img_load(x, b, ih0 + 2, iw0 + 1);   // k7
        s4 = 0.0f;                               // k pad
    }
    const v2f bB0 = { s0, s1 };
    const v2f bB1 = { s2, s3 };
    const v2f bB2 = { s4, 0.0f };

    const long outbase = (long)b * C_ * PPB;

    #pragma unroll 4
    for (int ct = 0; ct < 16; ++ct) {
        const int   ch = ct * 16 + n;            // lane's channel row (M = n)
        const float* wc = w + ch * 9;
        const v2f a0 = { wc[half ? 2 : 0], wc[half ? 3 : 1] };
        const v2f a1 = { wc[half ? 6 : 4], wc[half ? 7 : 5] };
        const v2f a2 = { half ? 0.0f : wc[8], 0.0f };

        v8f acc = {};
        acc = __builtin_amdgcn_wmma_f32_16x16x4_f32(false, a0, false, bB0,
                                                    (short)0, acc, false, false);
        acc = __builtin_amdgcn_wmma_f32_16x16x4_f32(false, a1, false, bB1,
                                                    (short)0, acc, false, false);
        acc = __builtin_amdgcn_wmma_f32_16x16x4_f32(false, a2, false, bB2,
                                                    (short)0, acc, false, false);

        const int chbase = ct * 16 + 8 * half;
        #pragma unroll
        for (int r = 0; r < 8; ++r) {
            // 16 contiguous floats per half-wave per VGPR row: coalesced.
            out[outbase + (long)(chbase + r) * PPB + p] = acc[r];
        }
    }
}

// ---------------------------------------------------------------------------
// Kernel 2: per-channel batch statistics, folded with gamma/beta into
// y = v*scale + shift.  One block per channel, tree-reduce in LDS.
// ---------------------------------------------------------------------------
__global__ void __launch_bounds__(256)
bn_stats_kernel(const float* __restrict__ h, float* __restrict__ stats,
                const float* __restrict__ gamma, const float* __restrict__ beta) {
    const int c = blockIdx.x;
    const int t = threadIdx.x;
    float sum = 0.0f, sq = 0.0f;
    for (int b = 0; b < B_; ++b) {
        const float* p = h + (long)b * C_ * PPB + (long)c * PPB;
        for (int i = t; i < PPB; i += 256) {
            float v = p[i];
            sum += v;
            sq  += v * v;
        }
    }
    __shared__ float ssum[256];
    __shared__ float ssq[256];
    ssum[t] = sum; ssq[t] = sq;
    __syncthreads();
    for (int s = 128; s > 0; s >>= 1) {
        if (t < s) { ssum[t] += ssum[t + s]; ssq[t] += ssq[t + s]; }
        __syncthreads();
    }
    if (t == 0) {
        const float N    = (float)(B_ * PPB);
        const float mean = ssum[0] / N;
        const float var  = ssq[0] / N - mean * mean;      // biased var (jnp.var)
        const float inv  = rsqrtf(var + 1e-5f);
        const float scale = inv * gamma[c];
        stats[c]        = scale;
        stats[C_ + c]   = beta[c] - mean * scale;
    }
}

// ---------------------------------------------------------------------------
// Kernel 3: in-place BN-apply + LIF scan over T=ow (40 steps) per (b,c,oh) row.
// v = v*0.5 + y; s = (v >= 1); v = s ? 0 : v.   Row = 40 contiguous floats.
// ---------------------------------------------------------------------------
__global__ void __launch_bounds__(256)
bn_lif_kernel(float* __restrict__ h, const float* __restrict__ stats) {
    const long tid = (long)blockIdx.x * blockDim.x + threadIdx.x; // b*C*OH rows
    const int  c   = (int)((tid >> 9) & 255);                     // (tid/512)%256
    const float scale = stats[c];
    const float shift = stats[C_ + c];
    float* p = h + tid * OW_;                                     // 160B aligned

    float v = 0.0f;
    #pragma unroll
    for (int q = 0; q < 10; ++q) {
        float4 xv = *(float4*)(p + q * 4);
        float y, s;
        y = xv.x * scale + shift; v = v * 0.5f + y; s = (v >= 1.0f) ? 1.0f : 0.0f; v = (s != 0.0f) ? 0.0f : v; xv.x = s;
        y = xv.y * scale + shift; v = v * 0.5f + y; s = (v >= 1.0f) ? 1.0f : 0.0f; v = (s != 0.0f) ? 0.0f : v; xv.y = s;
        y = xv.z * scale + shift; v = v * 0.5f + y; s = (v >= 1.0f) ? 1.0f : 0.0f; v = (s != 0.0f) ? 0.0f : v; xv.z = s;
        y = xv.w * scale + shift; v = v * 0.5f + y; s = (v >= 1.0f) ? 1.0f : 0.0f; v = (s != 0.0f) ? 0.0f : v; xv.w = s;
        *(float4*)(p + q * 4) = xv;
    }
}

extern "C" void kernel_launch(void* const* d_in, const int* in_sizes, int n_in,
                              void* d_out, int out_size, void* d_ws, size_t ws_size,
                              hipStream_t stream) {
    const float* x     = (const float*)d_in[0];   // (16, 80, 1024)
    const float* w     = (const float*)d_in[1];   // (256, 1, 3, 3)
    const float* gamma = (const float*)d_in[2];   // (256,)
    const float* beta  = (const float*)d_in[3];   // (256,)
    float* out   = (float*)d_out;                 // (16, 256*512, 40) fp32
    float* stats = (float*)d_ws;                  // 512 floats scratch

    // 20480 position-tiles total, 1 tile per wave, 8 waves per 256-thread block
    conv_wmma_kernel<<<(B_ * TPB_) / 8, 256, 0, stream>>>(x, w, out);
    bn_stats_kernel<<<C_, 256, 0, stream>>>(out, stats, gamma, beta);
    bn_lif_kernel<<<(B_ * C_ * OH_) / 256, 256, 0, stream>>>(out, stats);
}